// Optimized_Upsample_74818330296430
// MI455X (gfx1250) — compile-verified
//
#include <hip/hip_runtime.h>

typedef __attribute__((ext_vector_type(2))) float v2f;
typedef __attribute__((ext_vector_type(4))) float v4f;
typedef __attribute__((ext_vector_type(8))) float v8f;

#define B_  8
#define N_  16384
#define M_  1024
#define C_  256
#define PTS_PER_BLOCK 128   // 8 waves * 16 query points
#define NCHUNK (M_ / 16)    // 64 WMMA chunks per query tile

// Branchless sorted top-3 insert (ascending d0<=d1<=d2).
#define TOP3_INSERT(val, idx)                                              \
  {                                                                        \
    bool lt2 = (val) < d2;                                                 \
    bool lt1 = (val) < d1;                                                 \
    bool lt0 = (val) < d0;                                                 \
    float nd2 = lt1 ? d1 : (lt2 ? (val) : d2);                             \
    int   ni2 = lt1 ? i1 : (lt2 ? (idx) : i2);                             \
    float nd1 = lt0 ? d0 : (lt1 ? (val) : d1);                             \
    int   ni1 = lt0 ? i0 : (lt1 ? (idx) : i1);                             \
    float nd0 = lt0 ? (val) : d0;                                          \
    int   ni0 = lt0 ? (idx) : i0;                                          \
    d0 = nd0; d1 = nd1; d2 = nd2; i0 = ni0; i1 = ni1; i2 = ni2;            \
  }

// Phase 1: WMMA-based distance tiles + per-lane top-3 + IDW weights.
// Grid: B_ * (N_/PTS_PER_BLOCK) blocks of 256 threads (8 wave32).
__global__ __launch_bounds__(256) void knn_idw_phase1(
    const float* __restrict__ xyz,        // [B,N,3]
    const float* __restrict__ sxyz,       // [B,M,3]
    int*  __restrict__ wsIdx,             // [B*N,3]
    float* __restrict__ wsW)              // [B*N,3]
{
  __shared__ v4f sp[M_];                  // (x,y,z,||s||^2) per sampled point

  const int tid  = threadIdx.x;
  const int blocksPerBatch = N_ / PTS_PER_BLOCK;
  const int b     = blockIdx.x / blocksPerBatch;
  const int pbase = (blockIdx.x % blocksPerBatch) * PTS_PER_BLOCK;

  // Stage sampled points + norms into LDS (uniform trip count, no divergence).
  for (int m = tid; m < M_; m += 256) {
    const float* s = sxyz + ((size_t)b * M_ + m) * 3;
    float x = s[0], y = s[1], z = s[2];
    v4f p; p.x = x; p.y = y; p.z = z; p.w = x * x + y * y + z * z;
    sp[m] = p;
  }
  __syncthreads();

  const int wave = tid >> 5;
  const int lane = tid & 31;
  const int half = lane >> 4;   // 0: K rows {0,1} / D rows 0-7; 1: K {2,3} / D rows 8-15
  const int l16  = lane & 15;

  // This lane's query point (lane and lane^16 share the same query column).
  const int q = pbase + wave * 16 + l16;
  const float* qp = xyz + ((size_t)b * N_ + q) * 3;
  const float qx = qp[0], qy = qp[1], qz = qp[2];
  const float qn = qx * qx + qy * qy + qz * qz;

  // B fragment (4x16, constant across chunks):
  //   row0 = -2*qx, row1 = -2*qy, row2 = -2*qz, row3 = 1
  v2f bf;
  bf.x = half ? (-2.0f * qz) : (-2.0f * qx);
  bf.y = half ? 1.0f          : (-2.0f * qy);

  float d0 = 3.4e38f, d1 = 3.4e38f, d2 = 3.4e38f;
  int   i0 = 0, i1 = 0, i2 = 0;

  const float* spf = reinterpret_cast<const float*>(sp);
  const int laneOff = (l16 << 2) + (half << 1);   // float offset within a chunk row

  // Software pipeline: keep the A-fragment for chunk+1 in flight while the
  // top-3 insert chain (~100 dual-issued VALU ops) runs, so s_wait_dscnt at
  // the next WMMA finds the data already returned.
  v2f a;
  { const float* ap = spf + laneOff;  a.x = ap[0]; a.y = ap[1]; }

  for (int chunk = 0; chunk < NCHUNK; ++chunk) {
    v8f c = {0.f, 0.f, 0.f, 0.f, 0.f, 0.f, 0.f, 0.f};
    // D[m][n] = ||s_m||^2 - 2 s_m . q_n   (= dist^2 - ||q_n||^2, per-column const)
    v8f dacc = __builtin_amdgcn_wmma_f32_16x16x4_f32(
        false, a, false, bf, (short)0, c, false, false);

    // Prefetch next chunk's A fragment (clamped; last iteration reloads 63).
    {
      int nxt = chunk + 1 < NCHUNK ? chunk + 1 : NCHUNK - 1;
      const float* ap = spf + (nxt << 6) + laneOff;
      a.x = ap[0]; a.y = ap[1];
    }

    const int mbase = chunk * 16 + half * 8;
#pragma unroll
    for (int v = 0; v < 8; ++v) {
      float val = dacc[v];
      int   idx = mbase + v;
      TOP3_INSERT(val, idx);
    }
  }

  // Merge the two half-rows (lane n holds rows 0-7, lane n^16 holds rows 8-15).
  {
    float od0 = __shfl_xor(d0, 16, 32);
    float od1 = __shfl_xor(d1, 16, 32);
    float od2 = __shfl_xor(d2, 16, 32);
    int   oi0 = __shfl_xor(i0, 16, 32);
    int   oi1 = __shfl_xor(i1, 16, 32);
    int   oi2 = __shfl_xor(i2, 16, 32);
    TOP3_INSERT(od0, oi0);
    TOP3_INSERT(od1, oi1);
    TOP3_INSERT(od2, oi2);
  }

  if (half == 0) {
    // Recover true squared distances, clamp, IDW weights (P=2 on squared dist).
    float f0 = fmaxf(d0 + qn, 1e-10f);
    float f1 = fmaxf(d1 + qn, 1e-10f);
    float f2 = fmaxf(d2 + qn, 1e-10f);
    float w0 = 1.0f / (f0 * f0 + 1e-10f);
    float w1 = 1.0f / (f1 * f1 + 1e-10f);
    float w2 = 1.0f / (f2 * f2 + 1e-10f);
    float inv = 1.0f / (w0 + w1 + w2);

    const size_t row = (size_t)b * N_ + q;
    wsIdx[row * 3 + 0] = i0;
    wsIdx[row * 3 + 1] = i1;
    wsIdx[row * 3 + 2] = i2;
    wsW[row * 3 + 0] = w0 * inv;
    wsW[row * 3 + 1] = w1 * inv;
    wsW[row * 3 + 2] = w2 * inv;
  }
}

// Phase 2: streaming gather + blend. 32 lanes per output row, 8 channels/lane
// as two v4f's -> fully coalesced 1KB row writes. Feature gathers (8MB total)
// stay resident in the 192MB L2; output (128MB, write-once) is stored with a
// non-temporal hint so it never rinses the feature lines out of L2.
// Grid: (B_*N_)/8 blocks of 256 threads.
__global__ __launch_bounds__(256) void interp_phase2(
    const float* __restrict__ feats,      // [B,M,C]
    const int*  __restrict__ masks,       // [B,N]
    const int*  __restrict__ wsIdx,       // [B*N,3]
    const float* __restrict__ wsW,        // [B*N,3]
    float* __restrict__ out)              // [B,N,C]
{
  const int tid  = threadIdx.x;
  const int row  = blockIdx.x * 8 + (tid >> 5);
  const int lane = tid & 31;
  const int b    = row / N_;

  const float valid = masks[row] ? 1.0f : 0.0f;

  const int j0 = wsIdx[row * 3 + 0];
  const int j1 = wsIdx[row * 3 + 1];
  const int j2 = wsIdx[row * 3 + 2];
  const float w0 = wsW[row * 3 + 0] * valid;
  const float w1 = wsW[row * 3 + 1] * valid;
  const float w2 = wsW[row * 3 + 2] * valid;

  const v4f* F = reinterpret_cast<const v4f*>(feats) +
                 (size_t)b * M_ * (C_ / 4);
  const v4f* f0p = F + (size_t)j0 * (C_ / 4) + lane * 2;
  const v4f* f1p = F + (size_t)j1 * (C_ / 4) + lane * 2;
  const v4f* f2p = F + (size_t)j2 * (C_ / 4) + lane * 2;

  v4f a0 = f0p[0], a1 = f0p[1];
  v4f b0 = f1p[0], b1 = f1p[1];
  v4f c0 = f2p[0], c1 = f2p[1];

  v4f r0 = w0 * a0 + w1 * b0 + w2 * c0;
  v4f r1 = w0 * a1 + w1 * b1 + w2 * c1;

  v4f* O = reinterpret_cast<v4f*>(out) + (size_t)row * (C_ / 4) + lane * 2;
  __builtin_nontemporal_store(r0, O + 0);
  __builtin_nontemporal_store(r1, O + 1);
}

extern "C" void kernel_launch(void* const* d_in, const int* in_sizes, int n_in,
                              void* d_out, int out_size, void* d_ws, size_t ws_size,
                              hipStream_t stream) {
  const float* xyz   = (const float*)d_in[0];   // [B,N,3]
  const float* sxyz  = (const float*)d_in[1];   // [B,M,3]
  const float* feats = (const float*)d_in[2];   // [B,M,C]
  const int*   masks = (const int*)d_in[3];     // [B,N]
  float* out = (float*)d_out;

  // Workspace layout: [B*N*3] int indices, then [B*N*3] float weights (~3MB).
  int*   wsIdx = (int*)d_ws;
  float* wsW   = (float*)((char*)d_ws + (size_t)B_ * N_ * 3 * sizeof(int));

  dim3 blk(256);
  dim3 grid1(B_ * (N_ / PTS_PER_BLOCK));          // 1024 blocks
  knn_idw_phase1<<<grid1, blk, 0, stream>>>(xyz, sxyz, wsIdx, wsW);

  dim3 grid2((B_ * N_) / 8);                      // 16384 blocks
  interp_phase2<<<grid2, blk, 0, stream>>>(feats, masks, wsIdx, wsW, out);
}